// Model_74869869903920
// MI455X (gfx1250) — compile-verified
//
#include <hip/hip_runtime.h>
#include <hip/hip_bf16.h>

// GAT (2 layers) + classifier for MI455X / gfx1250.
//  - fp32 GEMM via V_WMMA_F32_16X16X4_F32, W quarter-panel resident in LDS
//  - A strips staged by the Tensor Data Mover (double-buffered, HW padding
//    reproduces the bank-conflict-free 260-float LDS row stride)
//  - wave-per-node attention kernel with global_prefetch of neighbor rows

typedef __attribute__((ext_vector_type(2))) float v2f;
typedef __attribute__((ext_vector_type(8))) float v8f;
typedef __attribute__((ext_vector_type(4))) unsigned int v4u;
typedef __attribute__((ext_vector_type(8))) int v8i;
typedef __attribute__((ext_vector_type(4))) int v4i;

#define DIMC   256
#define WCOLS  64       // quarter of W per workgroup
#define WSTR   72       // padded W row stride (72 % 64 == 8 -> conflict-free B frags)
#define ASTR   260      // padded A row stride (260 % 64 == 4 -> conflict-free A frags)
#define SROWS  32       // rows per staged A strip
#define GEMM_BLOCKS 160 // persistent blocks per quarter

// ---------------------------------------------------------------------------
// TDM: DMA one [rows x 256] fp32 strip at gbase into LDS with +4-DWORD row pad
// (6-arg toolchain variant: groups 0..3 + extra group + cpol)
// ---------------------------------------------------------------------------
__device__ __forceinline__ void tdm_load_strip(const float* gbase,
                                               unsigned lds_addr,
                                               int rows, int M) {
    const unsigned long long ga = (unsigned long long)(uintptr_t)gbase;
    v4u g0;
    g0[0] = 1u;                                           // count=1 (valid user D#)
    g0[1] = lds_addr;                                     // LDS byte address
    g0[2] = (unsigned)(ga & 0xFFFFFFFFull);               // global_addr[31:0]
    g0[3] = (unsigned)((ga >> 32) & 0x01FFFFFFull)        // global_addr[56:32]
          | 0x80000000u;                                  // type=2 ("image")
    v8i g1;
    g1[0] = (int)((2u << 16)      // data_size = 4B
          | (1u << 20)            // pad_enable
          | (7u << 22)            // pad_interval: every 256 DWORDs (one row)
          | (3u << 25));          // pad_amount: 4 DWORDs -> LDS stride 260
    g1[1] = (int)(256u << 16);                            // tensor_dim0 = 256
    g1[2] = (int)(((unsigned)M & 0xFFFFu) << 16);         // tensor_dim1 lo16
    g1[3] = (int)(((unsigned)M >> 16) | (256u << 16));    // dim1 hi | tile_dim0=256
    g1[4] = rows;                                         // tile_dim1 | tile_dim2=0
    g1[5] = 256;                                          // tensor_dim0_stride
    g1[6] = 0;
    g1[7] = 0;
    const v4i z4 = {0, 0, 0, 0};                          // 2D: groups 2/3 unused
    const v8i z8 = {0, 0, 0, 0, 0, 0, 0, 0};
#if __has_builtin(__builtin_amdgcn_tensor_load_to_lds)
    __builtin_amdgcn_tensor_load_to_lds(g0, g1, z4, z4, z8, 0);
#endif
}

// ---------------------------------------------------------------------------
// GEMM: out[M x 256] = A[M x 256] @ W[256 x 256]   (this block: 64 cols)
// ---------------------------------------------------------------------------
__global__ __launch_bounds__(256) void gat_gemm(const float* __restrict__ A,
                                                const float* __restrict__ W,
                                                float* __restrict__ out, int M) {
    __shared__ float Wlds[DIMC * WSTR];        // 72 KB: W[:, q*64 : q*64+64]
    __shared__ float Alds[2][SROWS * ASTR];    // 2 x 32.5 KB double-buffered strip

    const int tid     = threadIdx.x;
    const int wave    = tid >> 5;
    const int lane    = tid & 31;
    const int quarter = blockIdx.x & 3;
    const int blk     = blockIdx.x >> 2;
    const int nblk    = gridDim.x >> 2;

    // Stage the quarter weight panel once (float4, 16 iters/thread).
    for (int i = tid; i < DIMC * (WCOLS / 4); i += 256) {
        const int r = i >> 4, q = i & 15;
        ((float4*)&Wlds[r * WSTR])[q] =
            ((const float4*)(W + (size_t)r * DIMC + quarter * WCOLS))[q];
    }

    const int l15     = lane & 15;
    const int hsel    = lane >> 4;         // 0: K={0,1}/rows 0-7, 1: K={2,3}/rows 8-15
    const int n0      = (wave & 3) * 16;   // col tile within the 64-col panel
    const int rowhalf = wave >> 2;         // which 16-row half of the 32-row strip
    const int nStrips = (M + SROWS - 1) / SROWS;   // 1563 (tail = 16 rows)

    int buf = 0;
    if (wave == 0 && blk < nStrips) {
        const int m0 = blk * SROWS;
        tdm_load_strip(A + (size_t)m0 * DIMC,
                       (unsigned)(uintptr_t)&Alds[0][0],
                       min(SROWS, M - m0), M);
    }

    for (int s = blk; s < nStrips; s += nblk) {
        const int m0    = s * SROWS;
        const int snext = s + nblk;
        if (wave == 0) __builtin_amdgcn_s_wait_tensorcnt(0);
        __syncthreads();                   // strip `buf` visible to all waves
        if (wave == 0 && snext < nStrips) {
            const int mn = snext * SROWS;
            tdm_load_strip(A + (size_t)mn * DIMC,
                           (unsigned)(uintptr_t)&Alds[buf ^ 1][0],
                           min(SROWS, M - mn), M);
        }

        if (m0 + rowhalf * 16 < M) {       // uniform per wave: EXEC stays all-1s
            const float* __restrict__ Ab = &Alds[buf][0];
            v8f acc = {0.f, 0.f, 0.f, 0.f, 0.f, 0.f, 0.f, 0.f};
            const int arow = (rowhalf * 16 + l15) * ASTR;
            #pragma unroll 8
            for (int k0 = 0; k0 < DIMC; k0 += 4) {
                const int ka = k0 + 2 * hsel;
                v2f a, b;
                a[0] = Ab[arow + ka];              // A 16x4 fp32 fragment
                a[1] = Ab[arow + ka + 1];
                b[0] = Wlds[ka * WSTR + n0 + l15]; // B 4x16 fp32 fragment
                b[1] = Wlds[(ka + 1) * WSTR + n0 + l15];
                acc = __builtin_amdgcn_wmma_f32_16x16x4_f32(
                    false, a, false, b, (short)0, acc, false, false);
            }
            float* op = out + (size_t)(m0 + rowhalf * 16 + hsel * 8) * DIMC
                            + quarter * WCOLS + n0 + l15;
            #pragma unroll
            for (int r = 0; r < 8; ++r) op[(size_t)r * DIMC] = acc[r];
        }
        buf ^= 1;
    }
}

// ---------------------------------------------------------------------------
// Attention: one wave per node; lane owns channels d = lane + 32*j (coalesced)
// ---------------------------------------------------------------------------
__global__ __launch_bounds__(256) void gat_attn(const float* __restrict__ hw,
                                                const int* __restrict__ neigh,
                                                const float* __restrict__ a_s,
                                                const float* __restrict__ a_d,
                                                float* __restrict__ out, int N) {
    const int lane = threadIdx.x & 31;
    const int wave = threadIdx.x >> 5;
    const int node = blockIdx.x * 8 + wave;
    if (node >= N) return;

    // Load all 16 neighbor indices, then prefetch every neighbor row:
    // lanes cover the 8 cachelines (1 KB) of each row.
    const int* nb = neigh + (size_t)node * 16;
    int idx[16];
    #pragma unroll
    for (int k = 0; k < 16; ++k) idx[k] = nb[k];
    #pragma unroll
    for (int k = 0; k < 16; ++k)
        __builtin_prefetch(hw + (size_t)idx[k] * DIMC + (lane & 7) * 32, 0, 0);

    float asr[8], adr[8];
    #pragma unroll
    for (int j = 0; j < 8; ++j) {
        asr[j] = a_s[lane + 32 * j];
        adr[j] = a_d[lane + 32 * j];
    }

    // self term: (hw[n] . a_src), wave32 butterfly reduction
    const float* hn = hw + (size_t)node * DIMC;
    float p = 0.f;
    #pragma unroll
    for (int j = 0; j < 8; ++j) p = fmaf(hn[lane + 32 * j], asr[j], p);
    #pragma unroll
    for (int s = 16; s; s >>= 1) p += __shfl_xor(p, s, 32);

    float e[16];
    float nreg[16][8];                    // neighbor rows kept in VGPRs
    #pragma unroll
    for (int k = 0; k < 16; ++k) {
        const float* hk = hw + (size_t)idx[k] * DIMC;
        float q = 0.f;
        #pragma unroll
        for (int j = 0; j < 8; ++j) {
            const float v = hk[lane + 32 * j];
            nreg[k][j] = v;
            q = fmaf(v, adr[j], q);
        }
        #pragma unroll
        for (int s = 16; s; s >>= 1) q += __shfl_xor(q, s, 32);
        const float ek = p + q;
        e[k] = ek > 0.f ? ek : 0.2f * ek;        // LeakyReLU(0.2)
    }

    // softmax over 16 neighbors (replicated per lane)
    float m = e[0];
    #pragma unroll
    for (int k = 1; k < 16; ++k) m = fmaxf(m, e[k]);
    float sum = 0.f, wgt[16];
    #pragma unroll
    for (int k = 0; k < 16; ++k) { wgt[k] = __expf(e[k] - m); sum += wgt[k]; }
    const float inv = 1.f / sum;

    float o[8] = {0.f, 0.f, 0.f, 0.f, 0.f, 0.f, 0.f, 0.f};
    #pragma unroll
    for (int k = 0; k < 16; ++k) {
        const float a = wgt[k] * inv;
        #pragma unroll
        for (int j = 0; j < 8; ++j) o[j] = fmaf(a, nreg[k][j], o[j]);
    }

    float* on = out + (size_t)node * DIMC;
    #pragma unroll
    for (int j = 0; j < 8; ++j) {
        const float x = o[j];
        on[lane + 32 * j] = x > 0.f ? x : __expf(x) - 1.f;   // ELU
    }
}

// ---------------------------------------------------------------------------
// Classifier: block per batch item, wave per class group
// ---------------------------------------------------------------------------
__global__ __launch_bounds__(256) void gat_cls(const float* __restrict__ h,
                                               const int* __restrict__ node,
                                               const float* __restrict__ Wc,
                                               const float* __restrict__ bc,
                                               float* __restrict__ out,
                                               int B, int C) {
    const int b = blockIdx.x;
    if (b >= B) return;
    const int lane = threadIdx.x & 31;
    const int wave = threadIdx.x >> 5;

    const float* f = h + (size_t)node[b] * DIMC;
    float fr[8];
    #pragma unroll
    for (int j = 0; j < 8; ++j) fr[j] = f[lane + 32 * j];

    for (int c = wave; c < C; c += 8) {
        const float* w = Wc + (size_t)c * DIMC;
        float p = 0.f;
        #pragma unroll
        for (int j = 0; j < 8; ++j) p = fmaf(fr[j], w[lane + 32 * j], p);
        #pragma unroll
        for (int s = 16; s; s >>= 1) p += __shfl_xor(p, s, 32);
        if (lane == 0) out[(size_t)b * C + c] = p + bc[c];
    }
}

// ---------------------------------------------------------------------------
extern "C" void kernel_launch(void* const* d_in, const int* in_sizes, int n_in,
                              void* d_out, int out_size, void* d_ws, size_t ws_size,
                              hipStream_t stream) {
    const int N = 50000, D = 256, B = 4096, C = 50;

    const int*   node      = (const int*)d_in[0];
    const float* node_feat = (const float*)d_in[1];
    const int*   neigh     = (const int*)d_in[2];
    const float* Ws        = (const float*)d_in[3];   // [2][256][256]
    const float* a_src     = (const float*)d_in[4];   // [2][256]
    const float* a_dst     = (const float*)d_in[5];   // [2][256]
    const float* cls_W     = (const float*)d_in[6];   // [50][256]
    const float* cls_b     = (const float*)d_in[7];   // [50]
    float*       out       = (float*)d_out;

    float* hw = (float*)d_ws;                 // [N][D]  51.2 MB
    float* h  = hw + (size_t)N * D;           // [N][D]  51.2 MB

    const dim3 gGrid(4 * GEMM_BLOCKS);        // x4: one quarter of W per block
    const dim3 aGrid((N + 7) / 8);

    // layer 0
    gat_gemm<<<gGrid, 256, 0, stream>>>(node_feat, Ws, hw, N);
    gat_attn<<<aGrid, 256, 0, stream>>>(hw, neigh, a_src, a_dst, h, N);
    // layer 1 (h consumed by gemm before attn overwrites it)
    gat_gemm<<<gGrid, 256, 0, stream>>>(h, Ws + (size_t)D * D, hw, N);
    gat_attn<<<aGrid, 256, 0, stream>>>(hw, neigh, a_src + D, a_dst + D, h, N);
    // classifier
    gat_cls<<<B, 256, 0, stream>>>(h, node, cls_W, cls_b, out, B, C);
}